// Block_70257075028651
// MI455X (gfx1250) — compile-verified
//
#include <hip/hip_runtime.h>
#include <math.h>

typedef float v2f __attribute__((ext_vector_type(2)));
typedef float v8f __attribute__((ext_vector_type(8)));
typedef unsigned int u32x4 __attribute__((ext_vector_type(4)));
typedef int i32x4 __attribute__((ext_vector_type(4)));
typedef int i32x8 __attribute__((ext_vector_type(8)));

constexpr int kB = 2, kS = 1024, kD = 512, kH = 8, kHD = 64, kE = 8, kDF = 2048;
constexpr int kNT = kB * kS;                 // 2048 tokens
constexpr float kEps = 1e-5f;
constexpr float kScl = 0.04419417382415922f; // 1/sqrt(512)
constexpr int kFC = 256;                     // DF chunk staged in LDS
constexpr int kNC = kDF / kFC;               // 8 chunks
constexpr int kAP = 514;                     // padded LDS row stride for 512-f32 rows

__device__ __forceinline__ v8f wmma4(v2f a, v2f b, v8f c) {
  // V_WMMA_F32_16X16X4_F32 : D = A(16x4,f32) * B(4x16,f32) + C(16x16,f32)
  return __builtin_amdgcn_wmma_f32_16x16x4_f32(false, a, false, b, (short)0, c,
                                               false, false);
}

// Flat LDS pointer -> LDS byte offset (low 32 bits of the flat address)
__device__ __forceinline__ unsigned lds_off(const void* p) {
  return (unsigned)(unsigned long long)p;
}

// TDM: load a 16-row x 512-f32 tile (row stride 512 f32 in memory) into LDS,
// padding +1 DWORD after every 256 DWORDs => LDS row stride 514 f32.
__device__ __forceinline__ void tdm_load_16x512(const float* g, unsigned ldsaddr) {
  unsigned long long ga = (unsigned long long)(size_t)g;
  u32x4 g0;
  g0.x = 1u;                                              // count=1 (valid), user mode
  g0.y = ldsaddr;                                         // lds_addr (bytes)
  g0.z = (unsigned)(ga & 0xffffffffu);                    // global_addr[31:0]
  g0.w = (unsigned)((ga >> 32) & 0x01ffffffu) | (2u << 30); // addr[56:32] | type=2
  i32x8 g1;
  g1[0] = (int)((2u << 16) | (1u << 20) | (7u << 22));    // 4B elems, pad_en, every 256 DW, +1 DW
  g1[1] = (int)(512u << 16);                              // tensor_dim0 = 512 (lo16 @ [31:16])
  g1[2] = (int)(16u << 16);                               // dim0 hi=0 | tensor_dim1=16 (lo16)
  g1[3] = (int)(512u << 16);                              // dim1 hi=0 | tile_dim0=512
  g1[4] = (int)16;                                        // tile_dim1=16, tile_dim2=0
  g1[5] = (int)512;                                       // tensor_dim0_stride lo32 = 512
  g1[6] = 0;                                              // stride0 hi | dim1_stride lo
  g1[7] = 0;
  i32x4 z4 = {0, 0, 0, 0};
  i32x8 z8 = {0, 0, 0, 0, 0, 0, 0, 0};
#if __has_builtin(__builtin_amdgcn_tensor_load_to_lds)
  // 6-arg toolchain variant: (g0, g1, g2, g3, extra, cpol)
  __builtin_amdgcn_tensor_load_to_lds(g0, g1, z4, z4, z8, 0);
#else
  asm volatile("tensor_load_to_lds %0, %1" :: "s"(g0), "s"(g1) : "memory");
#endif
}

__device__ __forceinline__ void tdm_wait0() {
  __builtin_amdgcn_s_wait_tensorcnt(0);
}

// ---------------- LayerNorm: one wave per token ----------------
__global__ void ln_kernel(const float* __restrict__ x, const float* __restrict__ sc,
                          const float* __restrict__ bi, float* __restrict__ out) {
  int token = blockIdx.x * (blockDim.x >> 5) + (threadIdx.x >> 5);
  int lane = threadIdx.x & 31;
  const float* row = x + (size_t)token * kD;
  float v[16];
  float s = 0.f;
#pragma unroll
  for (int i = 0; i < 16; ++i) { v[i] = row[lane + 32 * i]; s += v[i]; }
#pragma unroll
  for (int o = 16; o >= 1; o >>= 1) s += __shfl_xor(s, o, 32);
  float mean = s * (1.0f / kD);
  float var = 0.f;
#pragma unroll
  for (int i = 0; i < 16; ++i) { float d = v[i] - mean; var += d * d; }
#pragma unroll
  for (int o = 16; o >= 1; o >>= 1) var += __shfl_xor(var, o, 32);
  float r = rsqrtf(var * (1.0f / kD) + kEps);
#pragma unroll
  for (int i = 0; i < 16; ++i) {
    int c = lane + 32 * i;
    out[(size_t)token * kD + c] = (v[i] - mean) * r * sc[c] + bi[c];
  }
}

// ------- QKV projection: block shares one TDM-staged A tile; wave = 16x16 tile -------
__global__ void qkv_kernel(const float* __restrict__ xln,
                           const float* __restrict__ Wq, const float* __restrict__ bq,
                           const float* __restrict__ Wk, const float* __restrict__ bk,
                           const float* __restrict__ Wv, const float* __restrict__ bv,
                           float* __restrict__ Q, float* __restrict__ K,
                           float* __restrict__ V) {
  __shared__ float As[16 * kAP];
  int wave = threadIdx.x >> 5, lane = threadIdx.x & 31;
  int half = lane >> 4, m = lane & 15;
  int mt = blockIdx.x;                         // 128 M-tiles
  if (threadIdx.x == 0) {
    tdm_load_16x512(xln + (size_t)mt * 16 * kD, lds_off(&As[0]));
    tdm_wait0();
  }
  __syncthreads();
  int task = blockIdx.y * 8 + wave;            // 0..95
  int nt = task & 3;                           // 4 N-tiles (HD=64)
  int h = (task >> 2) & 7;                     // head
  int mat = task >> 5;                         // 0=Q,1=K,2=V
  const float* W  = (mat == 0) ? Wq : (mat == 1) ? Wk : Wv;
  const float* bb = (mat == 0) ? bq : (mat == 1) ? bk : bv;
  float* O        = (mat == 0) ? Q  : (mat == 1) ? K  : V;
  const float* Bp = W + (size_t)h * kD * kHD + nt * 16;
  v8f c = {};
  for (int k = 0; k < kD; k += 4) {
    int kk = k + 2 * half;
    int kkp = kk + (kk >> 8);                  // skip the TDM pad DWORD
    v2f a = { As[m * kAP + kkp], As[m * kAP + kkp + 1] };
    v2f b = { Bp[(size_t)kk * kHD + m], Bp[(size_t)(kk + 1) * kHD + m] };
    c = wmma4(a, b, c);
  }
  float bias = bb[h * kHD + nt * 16 + m];
#pragma unroll
  for (int i = 0; i < 8; ++i) {
    int tok = mt * 16 + i + 8 * half;
    int b0 = tok >> 10, ss = tok & (kS - 1);
    O[(((size_t)(b0 * kH + h)) * kS + ss) * kHD + nt * 16 + m] = c[i] + bias;
  }
}

// ---------------- Flash attention: one wave per (b,h,16-query block) ------
__global__ void attn_kernel(const float* __restrict__ Q, const float* __restrict__ K,
                            const float* __restrict__ V, float* __restrict__ O) {
  __shared__ float Pl[8][16 * 17];
  int wave = threadIdx.x >> 5, lane = threadIdx.x & 31;
  int half = lane >> 4, m = lane & 15;
  int task = blockIdx.x * (blockDim.x >> 5) + wave;
  int qt = task & 63; int bh = task >> 6;
  int b0 = bh >> 3, h = bh & 7;
  const float* Qp = Q + ((size_t)(b0 * kH + h)) * kS * kHD;
  const float* Kp = K + ((size_t)(b0 * kH + h)) * kS * kHD;
  const float* Vp = V + ((size_t)(b0 * kH + h)) * kS * kHD;
  float* Ps = Pl[wave];

  v8f o0 = {}, o1 = {}, o2 = {}, o3 = {};
  float mrow[8], lrow[8];
#pragma unroll
  for (int i = 0; i < 8; ++i) { mrow[i] = -INFINITY; lrow[i] = 0.f; }

  for (int j = 0; j <= qt; ++j) {
    v8f cs = {};
    for (int k = 0; k < kHD; k += 4) {
      int kk = k + 2 * half;
      v2f a = { Qp[(qt * 16 + m) * kHD + kk], Qp[(qt * 16 + m) * kHD + kk + 1] };
      v2f b = { Kp[(j * 16 + m) * kHD + kk], Kp[(j * 16 + m) * kHD + kk + 1] };
      cs = wmma4(a, b, cs);
    }
#pragma unroll
    for (int i = 0; i < 8; ++i) {
      int r = i + 8 * half;
      float vsc = cs[i] * kScl;
      if (j == qt && m > r) vsc = -INFINITY;   // causal mask on diagonal tile
      float t = vsc;
#pragma unroll
      for (int o = 8; o >= 1; o >>= 1) t = fmaxf(t, __shfl_xor(t, o, 16));
      float mnew = fmaxf(mrow[i], t);
      float alpha = __expf(mrow[i] - mnew);
      float p = __expf(vsc - mnew);
      float rs = p;
#pragma unroll
      for (int o = 8; o >= 1; o >>= 1) rs += __shfl_xor(rs, o, 16);
      lrow[i] = lrow[i] * alpha + rs;
      mrow[i] = mnew;
      o0[i] *= alpha; o1[i] *= alpha; o2[i] *= alpha; o3[i] *= alpha;
      Ps[r * 17 + m] = p;                      // C-layout -> LDS (padded)
    }
    // P(16x16) @ V(16x64): A frags from LDS (transposed view), B from global
#pragma unroll
    for (int k = 0; k < 16; k += 4) {
      int kk = k + 2 * half;
      v2f a = { Ps[m * 17 + kk], Ps[m * 17 + kk + 1] };
      int v0 = (j * 16 + kk) * kHD, v1 = (j * 16 + kk + 1) * kHD;
      v2f b0 = { Vp[v0 + 0 * 16 + m], Vp[v1 + 0 * 16 + m] }; o0 = wmma4(a, b0, o0);
      v2f b1 = { Vp[v0 + 1 * 16 + m], Vp[v1 + 1 * 16 + m] }; o1 = wmma4(a, b1, o1);
      v2f b2 = { Vp[v0 + 2 * 16 + m], Vp[v1 + 2 * 16 + m] }; o2 = wmma4(a, b2, o2);
      v2f b3 = { Vp[v0 + 3 * 16 + m], Vp[v1 + 3 * 16 + m] }; o3 = wmma4(a, b3, o3);
    }
  }
#pragma unroll
  for (int i = 0; i < 8; ++i) {
    int s = qt * 16 + i + 8 * half;
    float inv = 1.0f / lrow[i];
    size_t base = ((size_t)(b0 * kS + s)) * kD + h * kHD + m;
    O[base + 0]  = o0[i] * inv;
    O[base + 16] = o1[i] * inv;
    O[base + 32] = o2[i] * inv;
    O[base + 48] = o3[i] * inv;
  }
}

// ------- Output projection + residual: TDM-staged A tile per block -------
__global__ void outproj_kernel(const float* __restrict__ Ao, const float* __restrict__ Wo,
                               const float* __restrict__ bo, const float* __restrict__ inp,
                               float* __restrict__ X) {
  __shared__ float As[16 * kAP];
  int wave = threadIdx.x >> 5, lane = threadIdx.x & 31;
  int half = lane >> 4, m = lane & 15;
  int mt = blockIdx.x;                         // 128 M-tiles
  if (threadIdx.x == 0) {
    tdm_load_16x512(Ao + (size_t)mt * 16 * kD, lds_off(&As[0]));
    tdm_wait0();
  }
  __syncthreads();
  int nt = blockIdx.y * 8 + wave;              // 32 N-tiles (D=512)
  const float* Bp = Wo + nt * 16;
  v8f c = {};
  for (int k = 0; k < kD; k += 4) {
    int kk = k + 2 * half;
    int kkp = kk + (kk >> 8);
    v2f a = { As[m * kAP + kkp], As[m * kAP + kkp + 1] };
    v2f b = { Bp[(size_t)kk * kD + m], Bp[(size_t)(kk + 1) * kD + m] };
    c = wmma4(a, b, c);
  }
  float bias = bo[nt * 16 + m];
#pragma unroll
  for (int i = 0; i < 8; ++i) {
    size_t idx = ((size_t)(mt * 16 + i + 8 * half)) * kD + nt * 16 + m;
    X[idx] = c[i] + bias + inp[idx];
  }
}

// ---------------- Router ----------------
__global__ void router_score_kernel(const float* __restrict__ y, const float* __restrict__ Wr,
                                    const float* __restrict__ br, float* __restrict__ rs) {
  int idx = blockIdx.x * blockDim.x + threadIdx.x;  // token*E + e
  if (idx >= kNT * kE) return;
  int tok = idx >> 3, e = idx & 7;
  const float* yr = y + (size_t)tok * kD;
  float s = 0.f;
  for (int d = 0; d < kD; ++d) s += yr[d] * Wr[d * kE + e];
  rs[idx] = s + br[e];
}

__global__ void router_topk_kernel(const float* __restrict__ rs, float* __restrict__ w) {
  int tok = blockIdx.x * blockDim.x + threadIdx.x;
  if (tok >= kNT) return;
  float sc[kE];
#pragma unroll
  for (int e = 0; e < kE; ++e) sc[e] = rs[tok * kE + e];
  int i0 = 0;
#pragma unroll
  for (int e = 1; e < kE; ++e) if (sc[e] > sc[i0]) i0 = e;   // earliest on ties
  int i1 = (i0 == 0) ? 1 : 0;
#pragma unroll
  for (int e = 0; e < kE; ++e) { if (e != i0 && sc[e] > sc[i1]) i1 = e; }
  float mx = fmaxf(sc[i0], sc[i1]);
  float e0 = __expf(sc[i0] - mx), e1 = __expf(sc[i1] - mx);
  float inv = 1.0f / (e0 + e1);
#pragma unroll
  for (int e = 0; e < kE; ++e) w[tok * kE + e] = 0.f;
  w[tok * kE + i0] = e0 * inv;
  w[tok * kE + i1] = e1 * inv;
}

// ------- MoE FFN: block = 16 tokens; Y tile TDM-staged once, reused by all experts -------
__global__ void moe_kernel(const float* __restrict__ y, const float* __restrict__ W1,
                           const float* __restrict__ b1, const float* __restrict__ W2,
                           const float* __restrict__ b2, const float* __restrict__ w,
                           const float* __restrict__ X, float* __restrict__ out) {
  __shared__ float Ys[16 * kAP];   // TDM-staged Y tile (padded rows)
  __shared__ float Hs[kFC * 17];   // transposed H chunk: Hs[f_local*17 + token]
  int wave = threadIdx.x >> 5, lane = threadIdx.x & 31;
  int half = lane >> 4, m = lane & 15;
  int mt = blockIdx.x;
  if (threadIdx.x == 0) {
    tdm_load_16x512(y + (size_t)mt * 16 * kD, lds_off(&Ys[0]));
    tdm_wait0();
  }
  __syncthreads();
  v8f acc[4] = {{}, {}, {}, {}};
  for (int e = 0; e < kE; ++e) {
    float wl = (lane < 16) ? w[(size_t)(mt * 16 + lane) * kE + e] : 0.f;
    if (__ballot(wl > 0.f) == 0) continue;   // uniform across block
    const float* W1e = W1 + (size_t)e * kD * kDF;
    const float* W2e = W2 + (size_t)e * kDF * kD;
    v8f eacc[4] = {{}, {}, {}, {}};
    for (int ch = 0; ch < kNC; ++ch) {
      int f0 = ch * kFC;
      if (ch + 1 < kNC) {                     // hint next weight chunk into L2
        __builtin_prefetch(&W1e[(size_t)(ch + 1) * kFC], 0, 1);
        __builtin_prefetch(&W2e[(size_t)((ch + 1) * kFC) * kD], 0, 1);
      }
      // phase 1: H = relu(Y @ W1 + b1) chunk -> LDS (A frags from TDM-staged Ys)
      for (int ft = 0; ft < kFC / 128; ++ft) {
        int fl = wave * (kFC / 8) + ft * 16;  // local f base
        int fc = f0 + fl;
        v8f c = {};
        for (int k = 0; k < kD; k += 4) {
          int kk = k + 2 * half;
          int kkp = kk + (kk >> 8);
          v2f a = { Ys[m * kAP + kkp], Ys[m * kAP + kkp + 1] };
          v2f b = { W1e[(size_t)kk * kDF + fc + m],
                    W1e[(size_t)(kk + 1) * kDF + fc + m] };
          c = wmma4(a, b, c);
        }
        float bias = b1[e * kDF + fc + m];
#pragma unroll
        for (int i = 0; i < 8; ++i)
          Hs[(fl + m) * 17 + i + 8 * half] = fmaxf(c[i] + bias, 0.f);
      }
      __syncthreads();
      // phase 2: EO += H @ W2 chunk
#pragma unroll
      for (int nt = 0; nt < 4; ++nt) {
        int ncol = wave * 64 + nt * 16;
        v8f c = eacc[nt];
        for (int k = 0; k < kFC; k += 4) {
          int kk = k + 2 * half;
          v2f a = { Hs[kk * 17 + m], Hs[(kk + 1) * 17 + m] };
          v2f b = { W2e[(size_t)(f0 + kk) * kD + ncol + m],
                    W2e[(size_t)(f0 + kk + 1) * kD + ncol + m] };
          c = wmma4(a, b, c);
        }
        eacc[nt] = c;
      }
      __syncthreads();
    }
#pragma unroll
    for (int nt = 0; nt < 4; ++nt) {
      float bias2 = b2[e * kD + wave * 64 + nt * 16 + m];
#pragma unroll
      for (int i = 0; i < 8; ++i) {
        int tok = mt * 16 + i + 8 * half;
        float we = w[(size_t)tok * kE + e];
        acc[nt][i] += we * (eacc[nt][i] + bias2);
      }
    }
  }
#pragma unroll
  for (int nt = 0; nt < 4; ++nt) {
#pragma unroll
    for (int i = 0; i < 8; ++i) {
      size_t idx = ((size_t)(mt * 16 + i + 8 * half)) * kD + wave * 64 + nt * 16 + m;
      out[idx] = acc[nt][i] + X[idx];
    }
  }
}

extern "C" void kernel_launch(void* const* d_in, const int* in_sizes, int n_in,
                              void* d_out, int out_size, void* d_ws, size_t ws_size,
                              hipStream_t stream) {
  (void)in_sizes; (void)n_in; (void)out_size; (void)ws_size;
  const float* inputs = (const float*)d_in[0];
  const float* ln1_s  = (const float*)d_in[1];
  const float* ln1_b  = (const float*)d_in[2];
  const float* Wq = (const float*)d_in[3];
  const float* bq = (const float*)d_in[4];
  const float* Wk = (const float*)d_in[5];
  const float* bk = (const float*)d_in[6];
  const float* Wv = (const float*)d_in[7];
  const float* bv = (const float*)d_in[8];
  const float* Wo = (const float*)d_in[9];
  const float* bo = (const float*)d_in[10];
  const float* ln2_s = (const float*)d_in[11];
  const float* ln2_b = (const float*)d_in[12];
  const float* Wr = (const float*)d_in[13];
  const float* br = (const float*)d_in[14];
  const float* W1 = (const float*)d_in[15];
  const float* b1 = (const float*)d_in[16];
  const float* W2 = (const float*)d_in[17];
  const float* b2 = (const float*)d_in[18];
  float* out = (float*)d_out;
  float* ws = (float*)d_ws;

  const size_t TD = (size_t)kNT * kD;   // 1,048,576 floats
  float* xln = ws;
  float* Qb  = ws + 1 * TD;
  float* Kb  = ws + 2 * TD;
  float* Vb  = ws + 3 * TD;
  float* ob  = ws + 4 * TD;
  float* xb  = ws + 5 * TD;
  float* yb  = ws + 6 * TD;
  float* rsb = ws + 7 * TD;
  float* wb  = rsb + (size_t)kNT * kE;

  ln_kernel<<<kNT / 8, 256, 0, stream>>>(inputs, ln1_s, ln1_b, xln);
  qkv_kernel<<<dim3(128, 12), 256, 0, stream>>>(xln, Wq, bq, Wk, bk, Wv, bv,
                                                Qb, Kb, Vb);
  attn_kernel<<<(kB * kH * (kS / 16)) / 8, 256, 0, stream>>>(Qb, Kb, Vb, ob);
  outproj_kernel<<<dim3(128, 4), 256, 0, stream>>>(ob, Wo, bo, inputs, xb);
  ln_kernel<<<kNT / 8, 256, 0, stream>>>(xb, ln2_s, ln2_b, yb);
  router_score_kernel<<<(kNT * kE + 255) / 256, 256, 0, stream>>>(yb, Wr, br, rsb);
  router_topk_kernel<<<(kNT + 255) / 256, 256, 0, stream>>>(rsb, wb);
  moe_kernel<<<kNT / 16, 256, 0, stream>>>(yb, W1, b1, W2, b2, wb, xb, out);
}